// KFiou_15762529976491
// MI455X (gfx1250) — compile-verified
//
#include <hip/hip_runtime.h>
#include <hip/hip_bf16.h>

#define T_BOXES   512
#define T_FLOATS  (T_BOXES * 5)          // 2560 floats per input per tile
#define NTHREADS  256
#define V2_PER_T  (T_FLOATS / 2 / NTHREADS)  // 5 float2 copies per thread per input

// ---------- async global->LDS path (gfx1250 ASYNCcnt) with safe fallback ----
#if defined(__has_builtin)
#if __has_builtin(__builtin_amdgcn_global_load_async_to_lds_b64)
#define HAVE_ASYNC 1
#endif
#endif
#ifndef HAVE_ASYNC
#define HAVE_ASYNC 0
#endif

#if HAVE_ASYNC
typedef int v2i __attribute__((vector_size(8)));            // 2 x i32
typedef __attribute__((address_space(1))) v2i* g_v2i_t;     // global
typedef __attribute__((address_space(3))) v2i* l_v2i_t;     // LDS
__device__ __forceinline__ void copy8(const float* g, float* l) {
  __builtin_amdgcn_global_load_async_to_lds_b64((g_v2i_t)g, (l_v2i_t)l, 0, 0);
}
#if __has_builtin(__builtin_amdgcn_s_wait_asynccnt)
#define WAIT_ASYNC(n) __builtin_amdgcn_s_wait_asynccnt(n)
#else
#define WAIT_ASYNC(n) asm volatile("s_wait_asynccnt " #n ::: "memory")
#endif
#else
__device__ __forceinline__ void copy8(const float* g, float* l) {
  *(float2*)l = *(const float2*)g;   // sync fallback, still coalesced b64
}
#define WAIT_ASYNC(n) ((void)0)
#endif

// ---------- math ------------------------------------------------------------
__device__ __forceinline__ void cov(float w, float h, float deg,
                                    float& s11, float& s12, float& s22) {
  w = fminf(fmaxf(w, 1e-7f), 1e7f);
  h = fminf(fmaxf(h, 1e-7f), 1e7f);
  float r = deg * (3.1415926f / 180.0f);
  float s, c;
  __sincosf(r, &s, &c);
  float a = 0.25f * w * w;   // (0.5 w)^2
  float b = 0.25f * h * h;
  s11 = a * c * c + b * s * s;
  s12 = (a - b) * s * c;
  s22 = a * s * s + b * c * c;
}

__device__ __forceinline__ float kfiou_one(float px, float py, float pw, float ph, float pr,
                                           float tx, float ty, float tw, float th_, float tr) {
  const float BETA = 1.0f / 9.0f;
  float dx = fabsf(px - tx), dy = fabsf(py - ty);
  float lx = (dx < BETA) ? (0.5f / BETA) * dx * dx : dx - 0.5f * BETA;
  float ly = (dy < BETA) ? (0.5f / BETA) * dy * dy : dy - 0.5f * BETA;
  float xy_loss = lx + ly;

  float p11, p12, p22, t11, t12, t22;
  cov(pw, ph, pr, p11, p12, p22);
  cov(tw, th_, tr, t11, t12, t22);

  float Vbp = 4.0f * sqrtf(p11 * p22 - p12 * p12);
  float Vbt = 4.0f * sqrtf(t11 * t22 - t12 * t12);

  float u11 = p11 + t11, u12 = p12 + t12, u22 = p22 + t22;
  float d = u11 * u22 - u12 * u12;
  float invd = 1.0f / d;                      // single divide, multiply through
  float K11 = (p11 * u22 - p12 * u12) * invd;
  float K12 = (p12 * u11 - p11 * u12) * invd;
  float K21 = (p12 * u22 - p22 * u12) * invd;
  float K22 = (p22 * u11 - p12 * u12) * invd;
  float S11 = p11 - (K11 * p11 + K12 * p12);
  float S12 = p12 - (K11 * p12 + K12 * p22);
  float S21 = p12 - (K21 * p11 + K22 * p12);
  float S22 = p22 - (K21 * p12 + K22 * p22);
  float Vb = 4.0f * sqrtf(S11 * S22 - S12 * S21);
  Vb = (Vb != Vb) ? 0.0f : Vb;                 // NaN -> 0
  float kf = Vb / (Vbp + Vbt - Vb + 1e-6f);
  return fmaxf(xy_loss + 1.0f - kf, 0.0f);
}

// ---------- kernel ----------------------------------------------------------
__global__ __launch_bounds__(NTHREADS)
void kfiou_kernel(const float* __restrict__ pred, const float* __restrict__ tgt,
                  float* __restrict__ out, int n) {
  __shared__ alignas(16) float lp[2][T_FLOATS];
  __shared__ alignas(16) float lt[2][T_FLOATS];

  const int tid    = threadIdx.x;
  const int nfull  = n / T_BOXES;
  const int stride = gridDim.x;

  auto issue_tile = [&](int tile, int buf) {
    const float* pg = pred + (size_t)tile * T_FLOATS;
    const float* tg = tgt  + (size_t)tile * T_FLOATS;
#pragma unroll
    for (int k = 0; k < V2_PER_T; ++k) {
      int f2 = tid + k * NTHREADS;            // float2 index in tile, fully coalesced
      copy8(pg + f2 * 2, &lp[buf][f2 * 2]);
      copy8(tg + f2 * 2, &lt[buf][f2 * 2]);
    }
  };

  if ((int)blockIdx.x < nfull) {
    int buf = 0;
    issue_tile(blockIdx.x, 0);                // prologue: fill buffer 0
    for (int t = blockIdx.x; t < nfull; t += stride) {
      int tn = t + stride;
      if (tn < nfull) {
        issue_tile(tn, buf ^ 1);              // prefetch next tile into other buffer
        WAIT_ASYNC(10);                       // retire current tile's 10 async b64 (in-order)
      } else {
        WAIT_ASYNC(0);
      }
      __syncthreads();                        // all waves' data for tile t landed
#pragma unroll
      for (int k = 0; k < T_BOXES / NTHREADS; ++k) {
        int bi = tid + k * NTHREADS;          // stride-5 LDS reads: conflict-free (gcd(5,64)=1)
        const float* p = &lp[buf][bi * 5];
        const float* q = &lt[buf][bi * 5];
        float v = kfiou_one(p[0], p[1], p[2], p[3], p[4],
                            q[0], q[1], q[2], q[3], q[4]);
        __builtin_nontemporal_store(v, &out[(size_t)t * T_BOXES + bi]);
      }
      __syncthreads();                        // done reading buf before it is refilled
      buf ^= 1;
    }
  }

  // tail tile (partial): one block, direct global reads (tiny fraction of work)
  int tail = n - nfull * T_BOXES;
  if (tail > 0 && (int)blockIdx.x == (nfull % stride)) {
    for (int i = tid; i < tail; i += NTHREADS) {
      size_t b = (size_t)nfull * T_BOXES + i;
      const float* p = pred + b * 5;
      const float* q = tgt  + b * 5;
      float v = kfiou_one(p[0], p[1], p[2], p[3], p[4],
                          q[0], q[1], q[2], q[3], q[4]);
      __builtin_nontemporal_store(v, &out[b]);
    }
  }
}

// ---------- launch ----------------------------------------------------------
extern "C" void kernel_launch(void* const* d_in, const int* in_sizes, int n_in,
                              void* d_out, int out_size, void* d_ws, size_t ws_size,
                              hipStream_t stream) {
  const float* pred = (const float*)d_in[0];
  const float* tgt  = (const float*)d_in[1];
  float* out = (float*)d_out;
  int n = in_sizes[0] / 5;                    // number of boxes

  int nfull = n / T_BOXES;
  int grid  = nfull < 1536 ? nfull : 1536;    // grid-stride; ~5 tiles/block at N=4M
  if (grid < 1) grid = 1;                     // still need a block for the tail

  kfiou_kernel<<<grid, NTHREADS, 0, stream>>>(pred, tgt, out, n);
}